// LlamaCrossAttention_62543313764835
// MI455X (gfx1250) — compile-verified
//
#include <hip/hip_runtime.h>
#include <hip/hip_bf16.h>
#include <math.h>

// ---- problem constants (match reference) ----
#define B_    16
#define HID_  1024
#define LH_   32
#define LD_   128
#define BLK_  16
#define NBPS_ 64
#define QDIM_ (LH_*LD_)   // 4096

typedef __attribute__((ext_vector_type(2))) float v2f;
typedef __attribute__((ext_vector_type(4))) float v4f;
typedef __attribute__((ext_vector_type(8))) float v8f;

__device__ __forceinline__ v2f nt_load2(const float* p) {
    return __builtin_nontemporal_load((const v2f*)p);
}
__device__ __forceinline__ v4f nt_load4(const float* p) {
    return __builtin_nontemporal_load((const v4f*)p);
}

// ============================================================================
// GEMM: C[16, NTOT] = A[16, K] * W[NTOT, K]^T   using V_WMMA_F32_16X16X4_F32.
// One 16-wide N tile per workgroup; K split over 4 waves, LDS tree reduce.
// A layout (16x4 f32): lane(0..15)=M, lane>>4 picks K pair {0,1}/{2,3};
// per-lane v2f = two consecutive K elements  -> b64 loads.
// B layout (4x16 f32): lane(0..15)=N col, same K-pair split -> rows of W are
// contiguous in K, also b64 loads.
// ============================================================================
template<int K, int NTOT>
__launch_bounds__(128)
__global__ void gemm16_wmma(const float* __restrict__ A,
                            const float* __restrict__ W,
                            float* __restrict__ C) {
    const int lane = threadIdx.x & 31;
    const int wv   = threadIdx.x >> 5;      // 0..3 (K-split)
    const int row  = lane & 15;             // M index for A / N col for B
    const int kh   = lane >> 4;             // which K pair
    const int n0   = blockIdx.x * 16;

    const float* Ap = A + row * K + 2 * kh;
    const float* Wp = W + (size_t)(n0 + row) * K + 2 * kh;

    v8f acc0 = (v8f)(0.0f);
    v8f acc1 = (v8f)(0.0f);

    const int kbeg = wv * (K / 4);
    const int kend = kbeg + (K / 4);
    #pragma unroll 4
    for (int k = kbeg; k < kend; k += 8) {
        v2f a0 = *(const v2f*)(Ap + k);
        v2f b0 = nt_load2(Wp + k);
        acc0 = __builtin_amdgcn_wmma_f32_16x16x4_f32(
                   false, a0, false, b0, (short)0, acc0, false, false);
        v2f a1 = *(const v2f*)(Ap + k + 4);
        v2f b1 = nt_load2(Wp + k + 4);
        acc1 = __builtin_amdgcn_wmma_f32_16x16x4_f32(
                   false, a1, false, b1, (short)0, acc1, false, false);
    }
    v8f acc = acc0 + acc1;

    __shared__ float red[4 * 32 * 8];
    #pragma unroll
    for (int r = 0; r < 8; r++) red[(wv * 32 + lane) * 8 + r] = acc[r];
    __syncthreads();

    if (threadIdx.x < 32) {
        #pragma unroll
        for (int r = 0; r < 8; r++) {
            float s = red[(0 * 32 + lane) * 8 + r] + red[(1 * 32 + lane) * 8 + r]
                    + red[(2 * 32 + lane) * 8 + r] + red[(3 * 32 + lane) * 8 + r];
            // C tile layout: VGPR r -> M = r + 8*kh, N = n0 + row
            const int m = r + 8 * kh;
            C[(size_t)m * NTOT + n0 + row] = s;
        }
    }
}

// ============================================================================
// Paged decode attention, one (batch, head) per workgroup, 8 waves.
// RoPE applied to q on load. KV streamed with non-temporal float4 loads
// (each (block, head) slice is read exactly once device-wide).
// ============================================================================
__launch_bounds__(256)
__global__ void paged_attn_kernel(const float* __restrict__ qraw,   // [B, LH*LD]
                                  const float* __restrict__ kc,     // [NBLK,LH,16,LD]
                                  const float* __restrict__ vc,
                                  const int*   __restrict__ btab,   // [B, NBPS]
                                  const int*   __restrict__ slen,   // [B]
                                  const int*   __restrict__ posid,  // [B]
                                  const int*   __restrict__ nspec,  // [1]
                                  float* __restrict__ attn_out) {   // [B, LH*LD]
    const int b    = blockIdx.x >> 5;
    const int h    = blockIdx.x & 31;
    const int tid  = threadIdx.x;
    const int lane = tid & 31;
    const int wv   = tid >> 5;           // 0..7

    const int   L   = slen[b];
    const float pos = (float)(posid[b] + nspec[0]);

    __shared__ float sc[NBPS_ * BLK_];   // scores / probabilities (4 KB)
    __shared__ float red[256];
    __shared__ float ored[8 * LD_];      // per-wave PV partials (4 KB)

    // ---- q load + RoPE: each lane owns dims d0..d0+3 (same map in all waves)
    const int d0 = lane * 4;
    const float* qb = qraw + (size_t)b * QDIM_ + h * LD_;
    v4f my  = *(const v4f*)(qb + d0);
    v4f oth = *(const v4f*)(qb + (d0 ^ 64));       // rotate-half partner
    const float sgn = (d0 < 64) ? -1.0f : 1.0f;
    v4f q4;
    #pragma unroll
    for (int i = 0; i < 4; i++) {
        const int   d   = d0 + i;
        const float fi  = (float)(d & 63);
        const float inv = __powf(10000.0f, -fi * (1.0f / 64.0f));
        float s, c;
        __sincosf(pos * inv, &s, &c);
        q4[i] = my[i] * c + sgn * oth[i] * s;
    }
    const float scale = 0.08838834764831845f;      // 1/sqrt(128)

    const int jmax = (L + BLK_ - 1) / BLK_;

    // ---- phase 1: scores  s[l] = (q . k_l) * scale
    for (int j = wv; j < jmax; j += 8) {
        const int blk = btab[b * NBPS_ + j];
        const float* kbase = kc + ((size_t)blk * LH_ + h) * (BLK_ * LD_);
        if (j + 8 < jmax) {
            const int nb = btab[b * NBPS_ + j + 8];
            __builtin_prefetch(kc + ((size_t)nb * LH_ + h) * (BLK_ * LD_), 0, 1);
        }
        const int rem  = L - j * BLK_;
        const int pmax = rem < BLK_ ? rem : BLK_;
        for (int p = 0; p < pmax; p++) {
            v4f kk = nt_load4(kbase + p * LD_ + d0);
            float part = kk[0]*q4[0] + kk[1]*q4[1] + kk[2]*q4[2] + kk[3]*q4[3];
            #pragma unroll
            for (int off = 16; off > 0; off >>= 1)
                part += __shfl_xor(part, off, 32);
            if (lane == 0) sc[j * BLK_ + p] = part * scale;
        }
    }
    __syncthreads();

    // ---- phase 2: softmax over sc[0..L)
    float lmax = -3.4028235e38f;
    for (int l = tid; l < L; l += 256) lmax = fmaxf(lmax, sc[l]);
    red[tid] = lmax; __syncthreads();
    for (int s = 128; s > 0; s >>= 1) {
        if (tid < s) red[tid] = fmaxf(red[tid], red[tid + s]);
        __syncthreads();
    }
    const float m = red[0];
    __syncthreads();

    float lsum = 0.0f;
    for (int l = tid; l < L; l += 256) {
        float e = __expf(sc[l] - m);
        sc[l] = e;                  // unnormalized prob
        lsum += e;
    }
    __syncthreads();
    red[tid] = lsum; __syncthreads();
    for (int s = 128; s > 0; s >>= 1) {
        if (tid < s) red[tid] += red[tid + s];
        __syncthreads();
    }
    const float invsum = 1.0f / red[0];

    // ---- phase 3: out[d] = sum_l p[l] * v[l][d]
    v4f acc = (v4f)(0.0f);
    for (int j = wv; j < jmax; j += 8) {
        const int blk = btab[b * NBPS_ + j];
        const float* vbase = vc + ((size_t)blk * LH_ + h) * (BLK_ * LD_);
        if (j + 8 < jmax) {
            const int nb = btab[b * NBPS_ + j + 8];
            __builtin_prefetch(vc + ((size_t)nb * LH_ + h) * (BLK_ * LD_), 0, 1);
        }
        const int rem  = L - j * BLK_;
        const int pmax = rem < BLK_ ? rem : BLK_;
        for (int p = 0; p < pmax; p++) {
            const float wp = sc[j * BLK_ + p];
            v4f vv = nt_load4(vbase + p * LD_ + d0);
            acc += vv * wp;
        }
    }
    *(v4f*)(ored + wv * LD_ + d0) = acc;
    __syncthreads();

    if (tid < LD_) {
        float s = 0.0f;
        #pragma unroll
        for (int w = 0; w < 8; w++) s += ored[w * LD_ + tid];
        attn_out[(size_t)b * QDIM_ + h * LD_ + tid] = s * invsum;
    }
}

// ============================================================================
extern "C" void kernel_launch(void* const* d_in, const int* in_sizes, int n_in,
                              void* d_out, int out_size, void* d_ws, size_t ws_size,
                              hipStream_t stream) {
    const float* hs    = (const float*)d_in[0];   // [16,1,1024]
    const float* wq    = (const float*)d_in[1];   // [4096,1024]
    const float* wo    = (const float*)d_in[2];   // [1024,4096]
    const float* kc    = (const float*)d_in[3];   // [1024,32,16,128]
    const float* vc    = (const float*)d_in[4];
    const int*   btab  = (const int*)d_in[5];     // [16,64]
    const int*   slen  = (const int*)d_in[6];     // [16]
    const int*   posid = (const int*)d_in[7];     // [16,1]
    const int*   nspec = (const int*)d_in[8];     // [1]
    float*       out   = (float*)d_out;           // [16,1,1024]

    float* ws_q    = (float*)d_ws;                // [16,4096] raw q
    float* ws_attn = ws_q + B_ * QDIM_;           // [16,4096] attn output

    // 1) q = hidden @ Wq^T      (WMMA f32 16x16x4)
    gemm16_wmma<HID_, QDIM_><<<QDIM_ / 16, 128, 0, stream>>>(hs, wq, ws_q);

    // 2) RoPE + paged attention (bandwidth-bound: NT streams of K/V)
    paged_attn_kernel<<<B_ * LH_, 256, 0, stream>>>(ws_q, kc, vc, btab, slen,
                                                    posid, nspec, ws_attn);

    // 3) out = attn @ Wo^T      (WMMA f32 16x16x4)
    gemm16_wmma<QDIM_, HID_><<<HID_ / 16, 128, 0, stream>>>(ws_attn, wo, out);
}